// GAT_87892210745357
// MI455X (gfx1250) — compile-verified
//
#include <hip/hip_runtime.h>
#include <hip/hip_bf16.h>

// ---------------------------------------------------------------------------
// GAT (2-layer, DGL-style GATConv) for MI455X / gfx1250.
// Dense projections use V_WMMA_F32_16X16X4_F32 (fp32 WMMA, wave32).
// Edge softmax / scatter use global float atomics (fits easily in 192MB L2).
// ---------------------------------------------------------------------------

#define GAT_H    4
#define GAT_F    64
#define GAT_HF   256   // H * F
#define NCLASSES 40

typedef float v2f __attribute__((ext_vector_type(2)));
typedef float v8f __attribute__((ext_vector_type(8)));

// ---------------- order-preserving float<->int for atomicMax --------------
__device__ __forceinline__ int f2ord(float f) {
    int b = __float_as_int(f);
    return b >= 0 ? b : (b ^ 0x7fffffff);
}
__device__ __forceinline__ float ord2f(int v) {
    int b = v >= 0 ? v : (v ^ 0x7fffffff);
    return __int_as_float(b);
}

// ---------------------------------------------------------------------------
// C[M x Nc] = A[M x K] @ B[K x Nc], fp32, one wave per 16x16 tile.
// Requires M % 16 == 0, Nc % 16 == 0, K % 4 == 0. blockDim.x = 128 (4 waves).
// Lane layout per CDNA5 ISA 7.12.2:
//   A 16x4 : lane l16,half h -> v0 = A[l16][k+2h], v1 = A[l16][k+2h+1]
//   B 4x16 : lane l16,half h -> v0 = B[k+2h][l16], v1 = B[k+2h+1][l16]
//   D 16x16: vgpr r, lane (l16 + 16h) -> C[r + 8h][l16]
// ---------------------------------------------------------------------------
__global__ void gemm_wmma_f32(const float* __restrict__ A,
                              const float* __restrict__ B,
                              float* __restrict__ C,
                              int M, int K, int Nc) {
    const int wave = blockIdx.x * (blockDim.x >> 5) + (threadIdx.x >> 5);
    const int lane = threadIdx.x & 31;
    const int ntn  = Nc >> 4;
    const int tm   = (wave / ntn) << 4;
    const int tn   = (wave % ntn) << 4;
    if (tm >= M) return;                    // wave-uniform branch

    const int half = lane >> 4;             // 0 or 1
    const int l16  = lane & 15;

    const float* arow = A + (size_t)(tm + l16) * K + 2 * half;
    const float* bcol = B + (size_t)(2 * half) * Nc + tn + l16;

    v8f acc = {};
    for (int k = 0; k < K; k += 4) {
        v2f a = *(const v2f*)arow;          // 8B aligned: row*K + 2h is even
        v2f b;
        b.x = bcol[0];
        b.y = bcol[Nc];
        acc = __builtin_amdgcn_wmma_f32_16x16x4_f32(
            /*neg_a=*/false, a, /*neg_b=*/false, b,
            /*c_mod=*/(short)0, acc, /*reuse_a=*/false, /*reuse_b=*/false);
        arow += 4;
        bcol += (size_t)4 * Nc;
    }

#pragma unroll
    for (int r = 0; r < 8; ++r) {
        C[(size_t)(tm + r + 8 * half) * Nc + tn + l16] = acc[r];
    }
}

// ---------------------------------------------------------------------------
// el[n,h] = dot(ft[n,h,:], al[h,:]);  er likewise.  One thread per (n,h).
// ft is row-major [N, H*F]; index i = n*H+h so ft + i*F is exactly ft[n,h,:].
// ---------------------------------------------------------------------------
__global__ void k_attn_coef(const float* __restrict__ ft,
                            const float* __restrict__ al,
                            const float* __restrict__ ar,
                            float* __restrict__ el,
                            float* __restrict__ er, int NH) {
    int i = blockIdx.x * blockDim.x + threadIdx.x;
    if (i >= NH) return;
    int h = i & (GAT_H - 1);
    const float* f = ft + (size_t)i * GAT_F;
    const float* a = al + h * GAT_F;
    const float* b = ar + h * GAT_F;
    float s0 = 0.f, s1 = 0.f;
#pragma unroll 8
    for (int j = 0; j < GAT_F; ++j) {
        float v = f[j];
        s0 += v * a[j];
        s1 += v * b[j];
    }
    el[i] = s0;
    er[i] = s1;
}

// ------------- init emax (ordered -inf), denom = 0, acc = 0 ----------------
__global__ void k_init(int* __restrict__ emaxi, float* __restrict__ denom,
                       float* __restrict__ acc, int NH, size_t accN) {
    size_t i = (size_t)blockIdx.x * blockDim.x + threadIdx.x;
    if (i < (size_t)NH) {
        emaxi[i] = f2ord(-INFINITY);
        denom[i] = 0.f;
    }
    if (i < accN) acc[i] = 0.f;
}

__device__ __forceinline__ float leaky02(float x) {
    return x > 0.f ? x : 0.2f * x;
}

// ------------- pass 1: segment max of leaky-relu logits over dst -----------
__global__ void k_edge_max(const int* __restrict__ src, const int* __restrict__ dst,
                           const float* __restrict__ el, const float* __restrict__ er,
                           int* __restrict__ emaxi, int EH) {
    int i = blockIdx.x * blockDim.x + threadIdx.x;
    if (i >= EH) return;
    int e = i >> 2, h = i & 3;
    int s = src[e], d = dst[e];
    float x = leaky02(el[s * GAT_H + h] + er[d * GAT_H + h]);
    atomicMax(&emaxi[d * GAT_H + h], f2ord(x));
}

// ------------- pass 2: a = exp(x - max); denom[dst] += a -------------------
__global__ void k_edge_exp(const int* __restrict__ src, const int* __restrict__ dst,
                           const float* __restrict__ el, const float* __restrict__ er,
                           const int* __restrict__ emaxi, float* __restrict__ denom,
                           float* __restrict__ aexp, int EH) {
    int i = blockIdx.x * blockDim.x + threadIdx.x;
    if (i >= EH) return;
    int e = i >> 2, h = i & 3;
    int s = src[e], d = dst[e];
    float x = leaky02(el[s * GAT_H + h] + er[d * GAT_H + h]);
    float m = ord2f(emaxi[d * GAT_H + h]);
    float a = __expf(x - m);
    aexp[i] = a;
    atomicAdd(&denom[d * GAT_H + h], a);
}

// ------------- pass 3: acc[dst,h,:] += (a/denom) * ft[src,h,:] -------------
// One wave per (edge, head); 32 lanes cover 64 features in 2 steps.
__global__ void k_edge_scatter(const int* __restrict__ src, const int* __restrict__ dst,
                               const float* __restrict__ aexp, const float* __restrict__ denom,
                               const float* __restrict__ ft, float* __restrict__ acc, int EH) {
    int wid  = blockIdx.x * (blockDim.x >> 5) + (threadIdx.x >> 5);
    int lane = threadIdx.x & 31;
    if (wid >= EH) return;
    int e = wid >> 2, h = wid & 3;
    int s = src[e], d = dst[e];
    float alpha = aexp[wid] / denom[d * GAT_H + h];
    const float* fs = ft  + (size_t)s * GAT_HF + h * GAT_F;
    float*       ad = acc + (size_t)d * GAT_HF + h * GAT_F;
    atomicAdd(&ad[lane],      alpha * fs[lane]);
    atomicAdd(&ad[lane + 32], alpha * fs[lane + 32]);
}

// ------------- bias + ELU, in place over [N, H*F] --------------------------
__global__ void k_bias_elu(float* __restrict__ acc, const float* __restrict__ bias,
                           size_t total) {
    size_t i = (size_t)blockIdx.x * blockDim.x + threadIdx.x;
    if (i >= total) return;
    float x = acc[i] + bias[i & (GAT_HF - 1)];
    acc[i] = x > 0.f ? x : (__expf(x) - 1.f);
}

// ------------- bias + ELU + mean over heads: [N,H*F] -> [N,F] --------------
__global__ void k_elu_mean(const float* __restrict__ acc, const float* __restrict__ bias,
                           float* __restrict__ hm, int N) {
    int i = blockIdx.x * blockDim.x + threadIdx.x;   // i = n*F + f
    if (i >= N * GAT_F) return;
    int n = i >> 6, f = i & (GAT_F - 1);
    float s = 0.f;
#pragma unroll
    for (int h = 0; h < GAT_H; ++h) {
        float x = acc[(size_t)n * GAT_HF + h * GAT_F + f] + bias[h * GAT_F + f];
        s += x > 0.f ? x : (__expf(x) - 1.f);
    }
    hm[i] = s * (1.0f / GAT_H);
}

// ------------- final classifier: out[n,c] = hm[n,:] @ fc_w[:,c] + fc_b -----
__global__ void k_fc(const float* __restrict__ hm, const float* __restrict__ w,
                     const float* __restrict__ b, float* __restrict__ out, int N) {
    int i = blockIdx.x * blockDim.x + threadIdx.x;   // i = n*NCLASSES + c
    if (i >= N * NCLASSES) return;
    int n = i / NCLASSES, c = i - n * NCLASSES;
    const float* x = hm + (size_t)n * GAT_F;
    float s = b[c];
#pragma unroll 8
    for (int f = 0; f < GAT_F; ++f) s += x[f] * w[f * NCLASSES + c];
    out[i] = s;
}

// ---------------------------------------------------------------------------
extern "C" void kernel_launch(void* const* d_in, const int* in_sizes, int n_in,
                              void* d_out, int out_size, void* d_ws, size_t ws_size,
                              hipStream_t stream) {
    const float* feat = (const float*)d_in[0];
    const int*   src  = (const int*)  d_in[1];
    const int*   dst  = (const int*)  d_in[2];
    const float* W1   = (const float*)d_in[3];
    const float* al1  = (const float*)d_in[4];
    const float* ar1  = (const float*)d_in[5];
    const float* b1   = (const float*)d_in[6];
    const float* W2   = (const float*)d_in[7];
    const float* al2  = (const float*)d_in[8];
    const float* ar2  = (const float*)d_in[9];
    const float* b2   = (const float*)d_in[10];
    const float* fcw  = (const float*)d_in[11];
    const float* fcb  = (const float*)d_in[12];
    float* out = (float*)d_out;

    const int IN_FEATS = 512;
    const int N  = in_sizes[0] / IN_FEATS;   // 20000
    const int E  = in_sizes[1];              // 320000
    const int NH = N * GAT_H;
    const int EH = E * GAT_H;
    const size_t NF = (size_t)N * GAT_HF;    // elements of an [N, H*F] buffer

    // ---- workspace arena (floats) ----
    float* buf0  = (float*)d_ws;             // layer1 ft, then layer2 acc
    float* buf1  = buf0 + NF;                // layer1 acc -> h1 (layer2 input)
    float* buf2  = buf1 + NF;                // layer2 ft
    float* el    = buf2 + NF;                // [N*H]
    float* er    = el + NH;
    int*   emaxi = (int*)(er + NH);          // [N*H]
    float* denom = (float*)(emaxi + NH);     // [N*H]
    float* aexp  = denom + NH;               // [E*H]
    float* hm    = aexp + EH;                // [N*F]

    const int TB = 256;
    dim3 blk(TB);
    auto cdiv = [](long long a, long long b) { return (int)((a + b - 1) / b); };

    const int gemm_waves = (N / 16) * (GAT_HF / 16);
    const int gemm_blocks = gemm_waves / 4;      // 4 waves (128 thr) per block
    const int scat_blocks = cdiv(EH, TB / 32);   // wave per (edge, head)

    // ================= layer 1 =================
    gemm_wmma_f32<<<gemm_blocks, 128, 0, stream>>>(feat, W1, buf0, N, IN_FEATS, GAT_HF);
    k_attn_coef<<<cdiv(NH, TB), blk, 0, stream>>>(buf0, al1, ar1, el, er, NH);
    k_init<<<cdiv((long long)NF, TB), blk, 0, stream>>>(emaxi, denom, buf1, NH, NF);
    k_edge_max<<<cdiv(EH, TB), blk, 0, stream>>>(src, dst, el, er, emaxi, EH);
    k_edge_exp<<<cdiv(EH, TB), blk, 0, stream>>>(src, dst, el, er, emaxi, denom, aexp, EH);
    k_edge_scatter<<<scat_blocks, blk, 0, stream>>>(src, dst, aexp, denom, buf0, buf1, EH);
    k_bias_elu<<<cdiv((long long)NF, TB), blk, 0, stream>>>(buf1, b1, NF);

    // ================= layer 2 =================
    gemm_wmma_f32<<<gemm_blocks, 128, 0, stream>>>(buf1, W2, buf2, N, GAT_HF, GAT_HF);
    k_attn_coef<<<cdiv(NH, TB), blk, 0, stream>>>(buf2, al2, ar2, el, er, NH);
    k_init<<<cdiv((long long)NF, TB), blk, 0, stream>>>(emaxi, denom, buf0, NH, NF);
    k_edge_max<<<cdiv(EH, TB), blk, 0, stream>>>(src, dst, el, er, emaxi, EH);
    k_edge_exp<<<cdiv(EH, TB), blk, 0, stream>>>(src, dst, el, er, emaxi, denom, aexp, EH);
    k_edge_scatter<<<scat_blocks, blk, 0, stream>>>(src, dst, aexp, denom, buf2, buf0, EH);

    // ============ head-mean + classifier =======
    k_elu_mean<<<cdiv(N * GAT_F, TB), blk, 0, stream>>>(buf0, b2, hm, N);
    k_fc<<<cdiv(N * NCLASSES, TB), blk, 0, stream>>>(hm, fcw, fcb, out, N);
}